// AIMNet2InteractionModule_52845277610673
// MI455X (gfx1250) — compile-verified
//
#include <hip/hip_runtime.h>
#include <hip/hip_bf16.h>

#define NP 160000
#define NA 10000
#define FDIM 256
#define GDIM 16
#define HDIM 64

typedef __attribute__((ext_vector_type(2))) float v2f;
typedef __attribute__((ext_vector_type(8))) float v8f;
typedef __attribute__((ext_vector_type(4))) unsigned int v4ui;
typedef __attribute__((ext_vector_type(4))) int v4si;
typedef __attribute__((ext_vector_type(8))) int v8si;

__device__ __forceinline__ v8f wmma_f32_k4(v2f a, v2f b, v8f c) {
  // V_WMMA_F32_16X16X4_F32: D = A(16x4) * B(4x16) + C(16x16), all fp32.
  return __builtin_amdgcn_wmma_f32_16x16x4_f32(false, a, false, b, (short)0, c,
                                               false, false);
}

__device__ __forceinline__ float gelu_exact(float x) {
  return 0.5f * x * (1.0f + erff(x * 0.70710678118654752f));
}

// TDM: async-copy a contiguous run of `elems` f32 from global to LDS.
// D# built per CDNA5 ISA ch.8 (group0/group1, 2D tile = 1 row x elems).
__device__ __forceinline__ void tdm_load_f32_row(unsigned int lds_off,
                                                 const float* gptr,
                                                 unsigned int elems) {
  unsigned long long ga = (unsigned long long)(uintptr_t)gptr;
  v4ui g0;
  g0.x = 1u;                                        // count=1 (valid user D#)
  g0.y = lds_off;                                   // lds_addr (bytes)
  g0.z = (unsigned int)ga;                          // global_addr[31:0]
  g0.w = (unsigned int)((ga >> 32) & 0x01FFFFFFu)   // global_addr[56:32]
         | (2u << 30);                              // type = 2 ("image")
  v8si g1;
  g1[0] = (int)(2u << 16);                          // data_size = 4B, mask=0
  g1[1] = (int)((elems & 0xFFFFu) << 16);           // tensor_dim0[15:0]
  g1[2] = (int)(((elems >> 16) & 0xFFFFu) | (1u << 16)); // dim0 hi | tensor_dim1=1
  g1[3] = (int)((elems & 0xFFFFu) << 16);           // tile_dim0 = elems
  g1[4] = 1;                                        // tile_dim1 = 1, tile_dim2 = 0
  g1[5] = (int)elems;                               // tensor_dim0_stride lo32
  g1[6] = 0;
  g1[7] = 0;
  v4si z4 = {0, 0, 0, 0};
#if __clang_major__ >= 23
  v8si z8 = {0, 0, 0, 0, 0, 0, 0, 0};
  __builtin_amdgcn_tensor_load_to_lds(g0, g1, z4, z4, z8, 0);
#else
  __builtin_amdgcn_tensor_load_to_lds(g0, g1, z4, z4, 0);
#endif
}

// ---------------------------------------------------------------------------
// Kernel 1: per-pair phase.
//   For a block of 32 pairs:
//     m(32x1024) = a_j(32x256) @ agh(256x1024)   [WMMA f32 16x16x4]
//     avf[p][h][d] += m[p][g*64+h] * gv[p][d][g]  (LDS f32 atomics)
//     global atomics: avf_v_sum, radial_emb, radial_q
// ---------------------------------------------------------------------------
__global__ __launch_bounds__(256) void aimnet_pair_kernel(
    const float* __restrict__ emb,      // (NA, 256)
    const float* __restrict__ q,        // (NA, 1)
    const int*   __restrict__ idxj,     // (NP)
    const float* __restrict__ gs,       // (NP, 16)
    const float* __restrict__ gv,       // (NP, 3, 16)
    const float* __restrict__ agh,      // (256, 1024) = (a, g*64+h)
    const float* __restrict__ Wgs,      // (256, 16)
    float* __restrict__ radial_emb,     // (NA, 256) accum
    float* __restrict__ radial_q,       // (NA, 256) accum
    float* __restrict__ avf)            // (NA, 64*3) accum, [atom][h*3+d]
{
  constexpr int SA_STRIDE = 260;                 // 256 + 4 pad (bank spread)
  __shared__ float sA[32 * SA_STRIDE];           // 33280 B, gathered a_j
  __shared__ float sAVF[32 * 192];               // 24576 B, [pair][h*3+d]
  __shared__ float sGVGS[32 * 48];               // 6144 B, gv then gs
  __shared__ int   sIdx[32];
  __shared__ float sQ[32];

  const int tid   = threadIdx.x;
  const int pbase = blockIdx.x * 32;
  const int lane  = tid & 31;
  const int w     = tid >> 5;     // wave id, 8 waves
  const int hi    = lane >> 4;    // 0/1: K-half / M-half selector
  const int lm    = lane & 15;

  // ---- phase 1a: indices + async TDM copy of gv block into LDS ----
  if (w == 0) {
    // Wave 0 issues the tensor DMA for the contiguous 32x48 f32 gv slab.
    tdm_load_f32_row((unsigned int)(uintptr_t)&sGVGS[0],
                     gv + (size_t)pbase * 48, 32u * 48u);
    sIdx[lane] = idxj[pbase + lane];
    sQ[lane]   = q[sIdx[lane]];
  }
  for (int i = tid; i < 32 * 192; i += 256) sAVF[i] = 0.0f;
  __syncthreads();

  // ---- phase 1b: gather a_j rows through sIdx ----
  for (int i = tid; i < 32 * FDIM; i += 256) {
    int row = i >> 8, f = i & 255;
    sA[row * SA_STRIDE + f] = emb[(size_t)sIdx[row] * FDIM + f];
  }
  if (w == 0) __builtin_amdgcn_s_wait_tensorcnt(0);   // gv DMA complete
  __syncthreads();

  // ---- phase 2: WMMA GEMM + fused gv contraction ----
  for (int i = 0; i < 8; ++i) {
    const int nt = w * 8 + i;             // N-tile 0..63, N = g*64 + h
    const int g  = nt >> 2;
    const int h0 = (nt & 3) << 4;
    const int n0 = nt << 4;
    v8f acc0 = {};                        // pairs 0..15
    v8f acc1 = {};                        // pairs 16..31
    const float* bp = agh + n0 + lm;
    #pragma unroll 4
    for (int k0 = 0; k0 < 256; k0 += 4) {
      const int kk = k0 + 2 * hi;
      v2f b;
      b.x = bp[kk * 1024];
      b.y = bp[(kk + 1) * 1024];
      v2f a0 = *(const v2f*)&sA[lm * SA_STRIDE + kk];
      v2f a1 = *(const v2f*)&sA[(16 + lm) * SA_STRIDE + kk];
      acc0 = wmma_f32_k4(a0, b, acc0);
      acc1 = wmma_f32_k4(a1, b, acc1);
    }
    // scatter: avf[p][h0+lm][d] += m * gv[p][d][g]
    // Batch the 6 gv broadcasts first so they clause; the non-returning
    // ds_add_f32 atomics then stream without dscnt waits.
    const int h = h0 + lm;
    #pragma unroll
    for (int r = 0; r < 8; ++r) {
      const int p0 = r + 8 * hi;
      const int p1 = 16 + p0;
      const float ga0 = sGVGS[p0 * 48 + g];
      const float ga1 = sGVGS[p0 * 48 + 16 + g];
      const float ga2 = sGVGS[p0 * 48 + 32 + g];
      const float gb0 = sGVGS[p1 * 48 + g];
      const float gb1 = sGVGS[p1 * 48 + 16 + g];
      const float gb2 = sGVGS[p1 * 48 + 32 + g];
      const float v0 = acc0[r];
      const float v1 = acc1[r];
      atomicAdd(&sAVF[p0 * 192 + h * 3 + 0], v0 * ga0);
      atomicAdd(&sAVF[p0 * 192 + h * 3 + 1], v0 * ga1);
      atomicAdd(&sAVF[p0 * 192 + h * 3 + 2], v0 * ga2);
      atomicAdd(&sAVF[p1 * 192 + h * 3 + 0], v1 * gb0);
      atomicAdd(&sAVF[p1 * 192 + h * 3 + 1], v1 * gb1);
      atomicAdd(&sAVF[p1 * 192 + h * 3 + 2], v1 * gb2);
    }
  }
  __syncthreads();

  // ---- phase 3: avf block-accum -> global segment sum ----
  for (int i = tid; i < 32 * 192; i += 256) {
    int row = i / 192, c = i - row * 192;
    atomicAdd(&avf[(size_t)sIdx[row] * 192 + c], sAVF[row * 192 + c]);
  }
  __syncthreads();

  // reuse sGVGS for raw gs
  for (int i = tid; i < 32 * 16; i += 256) {
    int row = i >> 4, gg = i & 15;
    sGVGS[row * 16 + gg] = gs[(size_t)(pbase + row) * 16 + gg];
  }
  __syncthreads();

  // ---- phase 4: mapped_gs radial / charge segment sums ----
  float wrow[16];
  #pragma unroll
  for (int gg = 0; gg < 16; ++gg) wrow[gg] = Wgs[tid * 16 + gg];  // tid == f
  for (int p = 0; p < 32; ++p) {
    float mp = 0.0f;
    #pragma unroll
    for (int gg = 0; gg < 16; ++gg) mp += sGVGS[p * 16 + gg] * wrow[gg];
    const int atom = sIdx[p];
    atomicAdd(&radial_emb[(size_t)atom * FDIM + tid], sA[p * SA_STRIDE + tid] * mp);
    atomicAdd(&radial_q[(size_t)atom * FDIM + tid], sQ[p] * mp);
  }
}

// ---------------------------------------------------------------------------
// Kernel 2: per-atom phase. Build combined(16x640), 3-layer MLP via WMMA.
// ---------------------------------------------------------------------------
__global__ __launch_bounds__(256) void aimnet_atom_kernel(
    const float* __restrict__ radial_emb,
    const float* __restrict__ radial_q,
    const float* __restrict__ avf,
    const float* __restrict__ W1, const float* __restrict__ b1,   // (256,640)
    const float* __restrict__ W2, const float* __restrict__ b2,   // (256,256)
    const float* __restrict__ W3, const float* __restrict__ b3,   // (258,256)
    float* __restrict__ out)   // [delta_a (NA*256) | delta_q (NA) | f (NA)]
{
  constexpr int IN_STRIDE = 644;               // 640 + 4 pad
  constexpr int H_STRIDE  = 260;               // 256 + 4 pad
  __shared__ float sIn[16 * IN_STRIDE];        // 41216 B
  __shared__ float sH [16 * H_STRIDE];         // 16640 B
  float* sH2 = sIn;                            // reuse sIn for layer-2 output

  const int tid   = threadIdx.x;
  const int abase = blockIdx.x * 16;
  const int lane = tid & 31, w = tid >> 5, hi = lane >> 4, lm = lane & 15;

  // ---- build combined = [radial_emb | ||avf|| | radial_q | zeros] ----
  for (int i = tid; i < 16 * 640; i += 256) {
    int m = i / 640, c = i - m * 640;
    int atom = abase + m;
    float v;
    if (c < 256) {
      v = radial_emb[(size_t)atom * 256 + c];
    } else if (c < 320) {
      int h = c - 256;
      float x = avf[(size_t)atom * 192 + h * 3 + 0];
      float y = avf[(size_t)atom * 192 + h * 3 + 1];
      float z = avf[(size_t)atom * 192 + h * 3 + 2];
      v = sqrtf(x * x + y * y + z * z);
    } else if (c < 576) {
      v = radial_q[(size_t)atom * 256 + (c - 320)];
    } else {
      v = 0.0f;
    }
    sIn[m * IN_STRIDE + c] = v;
  }
  __syncthreads();

  // ---- layer 1: gelu(combined @ W1^T + b1) -> sH ----
  for (int t = 0; t < 2; ++t) {
    const int nt = w * 2 + t;
    const int n  = nt * 16 + lm;
    v8f acc = {};
    #pragma unroll 4
    for (int k0 = 0; k0 < 640; k0 += 4) {
      const int kk = k0 + 2 * hi;
      v2f a = *(const v2f*)&sIn[lm * IN_STRIDE + kk];
      v2f b = *(const v2f*)&W1[n * 640 + kk];
      acc = wmma_f32_k4(a, b, acc);
    }
    const float bias = b1[n];
    #pragma unroll
    for (int r = 0; r < 8; ++r)
      sH[(r + 8 * hi) * H_STRIDE + n] = gelu_exact(acc[r] + bias);
  }
  __syncthreads();

  // ---- layer 2: gelu(h1 @ W2^T + b2) -> sH2 ----
  for (int t = 0; t < 2; ++t) {
    const int nt = w * 2 + t;
    const int n  = nt * 16 + lm;
    v8f acc = {};
    #pragma unroll 4
    for (int k0 = 0; k0 < 256; k0 += 4) {
      const int kk = k0 + 2 * hi;
      v2f a = *(const v2f*)&sH[lm * H_STRIDE + kk];
      v2f b = *(const v2f*)&W2[n * 256 + kk];
      acc = wmma_f32_k4(a, b, acc);
    }
    const float bias = b2[n];
    #pragma unroll
    for (int r = 0; r < 8; ++r)
      sH2[(r + 8 * hi) * H_STRIDE + n] = gelu_exact(acc[r] + bias);
  }
  __syncthreads();

  // ---- layer 3: h2 @ W3^T + b3, N = 258, scatter to output layout ----
  for (int nt = w; nt < 17; nt += 8) {      // wave-uniform trip count
    const int n = nt * 16 + lm;
    const bool nv = (n < 258);
    const v2f zero = {};
    v8f acc = {};
    #pragma unroll 4
    for (int k0 = 0; k0 < 256; k0 += 4) {
      const int kk = k0 + 2 * hi;
      v2f a = *(const v2f*)&sH2[lm * H_STRIDE + kk];
      v2f b = nv ? *(const v2f*)&W3[n * 256 + kk] : zero;
      acc = wmma_f32_k4(a, b, acc);
    }
    if (nv) {
      const float bias = b3[n];
      #pragma unroll
      for (int r = 0; r < 8; ++r) {
        const float x = acc[r] + bias;
        const int atom = abase + r + 8 * hi;
        if (n >= 2)       out[(size_t)atom * 256 + (n - 2)] = x;  // delta_a
        else if (n == 0)  out[(size_t)NA * 256 + atom]      = x;  // delta_q
        else              out[(size_t)NA * 256 + NA + atom] = x;  // f
      }
    }
  }
}

extern "C" void kernel_launch(void* const* d_in, const int* in_sizes, int n_in,
                              void* d_out, int out_size, void* d_ws, size_t ws_size,
                              hipStream_t stream) {
  const float* emb  = (const float*)d_in[0];
  const float* q    = (const float*)d_in[1];
  const int*   pair = (const int*)d_in[2];
  const int*   idxj = pair + NP;              // pair_indices[1]
  const float* gs   = (const float*)d_in[3];
  const float* gv   = (const float*)d_in[4];
  const float* agh  = (const float*)d_in[5];
  const float* Wgs  = (const float*)d_in[6];
  const float* W1   = (const float*)d_in[7];
  const float* b1   = (const float*)d_in[8];
  const float* W2   = (const float*)d_in[9];
  const float* b2   = (const float*)d_in[10];
  const float* W3   = (const float*)d_in[11];
  const float* b3   = (const float*)d_in[12];

  float* ws         = (float*)d_ws;
  float* radial_emb = ws;                         // NA*256
  float* radial_q   = ws + (size_t)NA * FDIM;     // NA*256
  float* avf        = ws + (size_t)2 * NA * FDIM; // NA*192

  const size_t accum_bytes =
      ((size_t)2 * NA * FDIM + (size_t)NA * 192) * sizeof(float);
  hipMemsetAsync(d_ws, 0, accum_bytes, stream);

  aimnet_pair_kernel<<<NP / 32, 256, 0, stream>>>(
      emb, q, idxj, gs, gv, agh, Wgs, radial_emb, radial_q, avf);

  aimnet_atom_kernel<<<NA / 16, 256, 0, stream>>>(
      radial_emb, radial_q, avf, W1, b1, W2, b2, W3, b3, (float*)d_out);
}